// DeepAttnBlock_81415400063710
// MI455X (gfx1250) — compile-verified
//
#include <hip/hip_runtime.h>
#include <math.h>

// Problem constants (match reference)
#define ND 128           // feature dim D
#define LEVELS 3
#define LN_EPS 1e-5f
#define SM_EPS 1e-16f
#define NEG_SLOPE 0.2f

typedef float v2f __attribute__((ext_vector_type(2)));
typedef float v8f __attribute__((ext_vector_type(8)));

// ---------------------------------------------------------------------------
// Per-level init: agg = 0, amax = -inf, denom = 0
// ---------------------------------------------------------------------------
__global__ __launch_bounds__(256) void init_level(float* __restrict__ agg,
                                                  unsigned* __restrict__ amax_bits,
                                                  float* __restrict__ denom, int n) {
    int i = blockIdx.x * blockDim.x + threadIdx.x;
    if (i < n * ND) agg[i] = 0.0f;
    if (i < n) { amax_bits[i] = 0xFF800000u; /* -inf */ denom[i] = 0.0f; }
}

// ---------------------------------------------------------------------------
// h = x @ W via V_WMMA_F32_16X16X4_F32 (full fp32, matches fp32 reference).
// One wave computes a 16x16 output tile; block of 256 = 8 waves covers the
// full 128-column output for a 16-row stripe. K=128 unrolled in steps of 4.
//
// fp32 16x16x4 fragment layout (wave32):
//   A (16x4):  lane<16 -> row=lane,    k={k0,k0+1};  lane>=16 -> row=lane-16, k={k0+2,k0+3}
//   B (4x16):  lane<16 -> col=lane,    k={k0,k0+1};  lane>=16 -> col=lane-16, k={k0+2,k0+3}
//   C (16x16): vgpr j: lanes0-15 -> M=j, lanes16-31 -> M=j+8; N=lane&15
// ---------------------------------------------------------------------------
__global__ __launch_bounds__(256) void gemm_wmma_f32(const float* __restrict__ X,
                                                     const float* __restrict__ Wm,
                                                     float* __restrict__ H, int n) {
    const int lane = threadIdx.x & 31;
    const int wave = threadIdx.x >> 5;
    const int half = lane >> 4;       // 0 or 1
    const int lr   = lane & 15;
    const int row0 = blockIdx.x * 16;
    const int col0 = wave * 16;
    if (row0 >= n) return;            // uniform per block (N % 16 == 0 anyway)

    v8f c = {0.f, 0.f, 0.f, 0.f, 0.f, 0.f, 0.f, 0.f};
    const float* __restrict__ xrow = X + (size_t)(row0 + lr) * ND;

#pragma unroll
    for (int k0 = 0; k0 < ND; k0 += 4) {
        v2f a = *(const v2f*)(xrow + k0 + 2 * half);              // contiguous 8B load
        v2f b;
        b.x = Wm[(size_t)(k0 + 2 * half + 0) * ND + col0 + lr];
        b.y = Wm[(size_t)(k0 + 2 * half + 1) * ND + col0 + lr];
        c = __builtin_amdgcn_wmma_f32_16x16x4_f32(false, a, false, b,
                                                  (short)0, c, false, false);
    }

#pragma unroll
    for (int j = 0; j < 8; ++j) {
        H[(size_t)(row0 + j + 8 * half) * ND + col0 + lr] = c[j];
    }
}

// ---------------------------------------------------------------------------
// Per-node attention logits: alpha_s[n] = h[n].a_src, alpha_d[n] = h[n].a_dst
// One wave per node; lane handles 4 contiguous floats (float4), wave-reduce.
// ---------------------------------------------------------------------------
__global__ __launch_bounds__(256) void node_alpha(const float* __restrict__ H,
                                                  const float* __restrict__ asrc,
                                                  const float* __restrict__ adst,
                                                  float* __restrict__ out_s,
                                                  float* __restrict__ out_d, int n) {
    const int lane = threadIdx.x & 31;
    const int node = blockIdx.x * (blockDim.x >> 5) + (threadIdx.x >> 5);
    if (node >= n) return;
    const float4 hv = ((const float4*)(H + (size_t)node * ND))[lane];
    const float4 as = ((const float4*)asrc)[lane];
    const float4 ad = ((const float4*)adst)[lane];
    float ps = hv.x * as.x + hv.y * as.y + hv.z * as.z + hv.w * as.w;
    float pd = hv.x * ad.x + hv.y * ad.y + hv.z * ad.z + hv.w * ad.w;
#pragma unroll
    for (int off = 16; off > 0; off >>= 1) {
        ps += __shfl_xor(ps, off, 32);
        pd += __shfl_xor(pd, off, 32);
    }
    if (lane == 0) { out_s[node] = ps; out_d[node] = pd; }
}

// ---------------------------------------------------------------------------
// Edge helpers
// ---------------------------------------------------------------------------
__device__ __forceinline__ void edge_sd(const int* __restrict__ ei, int e, int E,
                                        int& s, int& d) {
    if (e < E) { s = ei[e]; d = ei[E + e]; }
    else       { s = e - E; d = e - E; }       // appended self-loops
}

__device__ __forceinline__ float lrelu(float z) { return z > 0.0f ? z : NEG_SLOPE * z; }

// Float atomic-max via signed/unsigned bit trick (valid for IEEE-754 ordering;
// amax initialized to -inf bit pattern 0xFF800000).
__device__ __forceinline__ void atomic_max_f32(float* addr, float val) {
    if (val >= 0.0f) atomicMax((int*)addr, __float_as_int(val));
    else             atomicMin((unsigned int*)addr, __float_as_uint(val));
}

// Pass 1: segment max of edge logits over dst
__global__ __launch_bounds__(256) void edge_max(const int* __restrict__ ei,
                                                const float* __restrict__ as_,
                                                const float* __restrict__ ad_,
                                                float* __restrict__ amax,
                                                int E, int E2) {
    int e = blockIdx.x * blockDim.x + threadIdx.x;
    if (e >= E2) return;
    int s, d; edge_sd(ei, e, E, s, d);
    atomic_max_f32(&amax[d], lrelu(as_[s] + ad_[d]));
}

// Pass 2: denom[dst] += exp(z - amax[dst])
__global__ __launch_bounds__(256) void edge_expsum(const int* __restrict__ ei,
                                                   const float* __restrict__ as_,
                                                   const float* __restrict__ ad_,
                                                   const float* __restrict__ amax,
                                                   float* __restrict__ denom,
                                                   int E, int E2) {
    int e = blockIdx.x * blockDim.x + threadIdx.x;
    if (e >= E2) return;
    int s, d; edge_sd(ei, e, E, s, d);
    float p = __expf(lrelu(as_[s] + ad_[d]) - amax[d]);
    atomicAdd(&denom[d], p);
}

// Pass 3: agg[dst,:] += h[src,:] * (exp(z - amax[dst]) / (denom[dst] + eps))
// One wave per edge; lane gathers a float4 of h[src] and does 4 f32 atomics.
// L2-resident: h (51MB) + agg (51MB) both fit in the 192MB L2.
__global__ __launch_bounds__(256) void edge_aggregate(const int* __restrict__ ei,
                                                      const float* __restrict__ H,
                                                      const float* __restrict__ as_,
                                                      const float* __restrict__ ad_,
                                                      const float* __restrict__ amax,
                                                      const float* __restrict__ denom,
                                                      float* __restrict__ agg,
                                                      int E, int E2) {
    const int lane = threadIdx.x & 31;
    const int e = blockIdx.x * (blockDim.x >> 5) + (threadIdx.x >> 5);
    if (e >= E2) return;
    int s, d; edge_sd(ei, e, E, s, d);
    const float p = __expf(lrelu(as_[s] + ad_[d]) - amax[d]);
    const float w = p / (denom[d] + SM_EPS);
    const float4 hv = ((const float4*)(H + (size_t)s * ND))[lane];
    float* dst = agg + (size_t)d * ND + lane * 4;
    atomicAdd(dst + 0, hv.x * w);
    atomicAdd(dst + 1, hv.y * w);
    atomicAdd(dst + 2, hv.z * w);
    atomicAdd(dst + 3, hv.w * w);
}

// ---------------------------------------------------------------------------
// y = x + agg + bias; LayerNorm(y)*gamma + beta; ReLU -> xout
// One wave per node, lane handles 4 floats, two wave reductions (mean, var).
// ---------------------------------------------------------------------------
__global__ __launch_bounds__(256) void finalize_node(const float* __restrict__ Xin,
                                                     const float* __restrict__ agg,
                                                     const float* __restrict__ bias,
                                                     const float* __restrict__ gamma,
                                                     const float* __restrict__ beta,
                                                     float* __restrict__ Xout, int n) {
    const int lane = threadIdx.x & 31;
    const int node = blockIdx.x * (blockDim.x >> 5) + (threadIdx.x >> 5);
    if (node >= n) return;
    const float4 xv = ((const float4*)(Xin + (size_t)node * ND))[lane];
    const float4 av = ((const float4*)(agg + (size_t)node * ND))[lane];
    const float4 bv = ((const float4*)bias)[lane];
    float4 y;
    y.x = xv.x + av.x + bv.x;
    y.y = xv.y + av.y + bv.y;
    y.z = xv.z + av.z + bv.z;
    y.w = xv.w + av.w + bv.w;

    float s = y.x + y.y + y.z + y.w;
#pragma unroll
    for (int off = 16; off > 0; off >>= 1) s += __shfl_xor(s, off, 32);
    const float mu = s * (1.0f / ND);

    float q = (y.x - mu) * (y.x - mu) + (y.y - mu) * (y.y - mu) +
              (y.z - mu) * (y.z - mu) + (y.w - mu) * (y.w - mu);
#pragma unroll
    for (int off = 16; off > 0; off >>= 1) q += __shfl_xor(q, off, 32);
    const float inv = rsqrtf(q * (1.0f / ND) + LN_EPS);

    const float4 gv = ((const float4*)gamma)[lane];
    const float4 tv = ((const float4*)beta)[lane];
    float4 o;
    o.x = fmaxf(0.0f, (y.x - mu) * inv * gv.x + tv.x);
    o.y = fmaxf(0.0f, (y.y - mu) * inv * gv.y + tv.y);
    o.z = fmaxf(0.0f, (y.z - mu) * inv * gv.z + tv.z);
    o.w = fmaxf(0.0f, (y.w - mu) * inv * gv.w + tv.w);
    ((float4*)(Xout + (size_t)node * ND))[lane] = o;
}

// ---------------------------------------------------------------------------
// Host-side launch
// ---------------------------------------------------------------------------
extern "C" void kernel_launch(void* const* d_in, const int* in_sizes, int n_in,
                              void* d_out, int out_size, void* d_ws, size_t ws_size,
                              hipStream_t stream) {
    const float* x        = (const float*)d_in[0];   // [N, 128]
    const int*   ei       = (const int*)d_in[1];     // [2, E] flattened
    const float* W        = (const float*)d_in[2];   // [3, 128, 128]
    const float* att_src  = (const float*)d_in[3];   // [3, 128]
    const float* att_dst  = (const float*)d_in[4];   // [3, 128]
    const float* bias     = (const float*)d_in[5];   // [3, 128]
    const float* ln_gamma = (const float*)d_in[6];   // [3, 128]
    const float* ln_beta  = (const float*)d_in[7];   // [3, 128]

    const int n  = in_sizes[0] / ND;   // 100000
    const int E  = in_sizes[1] / 2;    // 1600000
    const int E2 = E + n;              // with self-loops

    // Workspace layout (floats): h[N*D] | agg[N*D] | as[N] | ad[N] | amax[N] | denom[N]
    float* ws    = (float*)d_ws;
    float* h     = ws;
    float* agg   = h + (size_t)n * ND;
    float* as_   = agg + (size_t)n * ND;
    float* ad_   = as_ + n;
    float* amax  = ad_ + n;
    float* denom = amax + n;
    float* xout  = (float*)d_out;

    const int waves_per_block = 8;     // 256 threads

    for (int l = 0; l < LEVELS; ++l) {
        const float* xin = (l == 0) ? x : (const float*)xout;

        init_level<<<(n * ND + 255) / 256, 256, 0, stream>>>(agg, (unsigned*)amax, denom, n);

        gemm_wmma_f32<<<(n + 15) / 16, 256, 0, stream>>>(
            xin, W + (size_t)l * ND * ND, h, n);

        node_alpha<<<(n + waves_per_block - 1) / waves_per_block, 256, 0, stream>>>(
            h, att_src + l * ND, att_dst + l * ND, as_, ad_, n);

        edge_max<<<(E2 + 255) / 256, 256, 0, stream>>>(ei, as_, ad_, amax, E, E2);

        edge_expsum<<<(E2 + 255) / 256, 256, 0, stream>>>(ei, as_, ad_, amax, denom, E, E2);

        edge_aggregate<<<(E2 + waves_per_block - 1) / waves_per_block, 256, 0, stream>>>(
            ei, h, as_, ad_, amax, denom, agg, E, E2);

        finalize_node<<<(n + waves_per_block - 1) / waves_per_block, 256, 0, stream>>>(
            xin, agg, bias + l * ND, ln_gamma + l * ND, ln_beta + l * ND, xout, n);
    }
}